// VRPPolarEdgeEmbedding_86174223827068
// MI455X (gfx1250) — compile-verified
//
#include <hip/hip_runtime.h>

typedef __attribute__((ext_vector_type(2))) float v2f;
typedef __attribute__((ext_vector_type(4))) float v4f;
typedef __attribute__((ext_vector_type(8))) float v8f;

#define NB     16
#define NN     1000
#define KSP    100
#define ESZ    (KSP + NN * KSP)   // 100100 edges per batch
#define SORT_N 1024
#define BLK    256
#define DEMB   128
#define ROWPAD 112                // 100 edge rows padded to 7 M-tiles

// d_out flat layout (float32), reference return order:
//   init_embeddings [16,1000,128] | edge_index [16,2,E] | emb [16,E,128] | mask [16,E]
#define OFF_EIDX 2048000ULL
#define OFF_EMB  5251200ULL
#define OFF_MASK 210256000ULL

__global__ __launch_bounds__(BLK) void vrp_edges_kernel(
    const float* __restrict__ locs,     // [16,1000,2]
    const float* __restrict__ edge_w,   // [2,128]
    const float* __restrict__ edge_b,   // [128]
    float* __restrict__ out) {
  const int  b      = blockIdx.x / (NN + 1);
  const int  r      = blockIdx.x % (NN + 1);
  const bool depot  = (r == 0);
  const int  srcRow = depot ? 0 : (r - 1);
  const float* L = locs + (size_t)b * NN * 2;

  // Overlay: sort keys (8KB) are dead before the emb tile (57KB) is written.
  __shared__ __align__(16) unsigned char shraw[ROWPAD * DEMB * sizeof(float)];
  unsigned long long* keys = reinterpret_cast<unsigned long long*>(shraw);
  float* embS = reinterpret_cast<float*>(shraw);
  __shared__ v2f   a12s[ROWPAD];   // (a1, a2) interleaved -> single ds_load_b64
  __shared__ float mskf[ROWPAD];   // 1.0 / 0.0 per edge row

  const int tid = threadIdx.x;
  const float xd = L[0], yd = L[1];                   // depot location
  const float x0 = L[2 * srcRow], y0 = L[2 * srcRow + 1];
  const float thI = atan2f(y0 - yd, x0 - xd);         // theta of source row

  // ---- build composite sort keys: value = a1 (node rows) or dist (depot row)
  for (int j = tid; j < SORT_N; j += BLK) {
    unsigned long long key = ~0ULL;                   // pad slots sort last
    if (j < NN) {
      float val;
      if (depot) {
        float dx = L[2 * j] - x0, dy = L[2 * j + 1] - y0;
        val = sqrtf(dx * dx + dy * dy);
      } else if (j == srcRow) {
        val = __builtin_inff();                       // exclude diagonal
      } else {
        float thJ = atan2f(L[2 * j + 1] - yd, L[2 * j] - xd);
        val = 1.0f - cosf(thI - thJ);
      }
      // val >= 0 so float bits are order-preserving; low 32 bits = index
      key = ((unsigned long long)__float_as_uint(val) << 32) | (unsigned int)j;
    }
    keys[j] = key;
  }
  __syncthreads();

  // ---- bitonic sort ascending (matches lax.top_k tie-break: lower index first)
  for (int k = 2; k <= SORT_N; k <<= 1) {
    for (int j = k >> 1; j > 0; j >>= 1) {
      for (int i = tid; i < SORT_N; i += BLK) {
        int ixj = i ^ j;
        if (ixj > i) {
          bool up = ((i & k) == 0);
          unsigned long long a = keys[i], c = keys[ixj];
          if ((a > c) == up) { keys[i] = c; keys[ixj] = a; }
        }
      }
      __syncthreads();
    }
  }

  // ---- emit edge_index / mask, stage attrs for WMMA (pad 100 -> 112 rows)
  const size_t e0 = depot ? 0u : (size_t)(KSP + srcRow * KSP);
  if (tid < ROWPAD) {
    if (tid < KSP) {
      unsigned long long key = keys[tid];
      int   d   = (int)(unsigned int)(key & 0xffffffffULL);
      float val = __uint_as_float((unsigned int)(key >> 32));
      float a1, a2;
      if (depot) {
        a2 = val;
        float thD = atan2f(L[2 * d + 1] - yd, L[2 * d] - xd);
        a1 = 1.0f - cosf(thI - thD);
      } else {
        a1 = val;
        float dx = L[2 * d] - x0, dy = L[2 * d + 1] - y0;
        a2 = sqrtf(dx * dx + dy * dy);
      }
      bool m = (d != 0) && (depot || srcRow != 0);
      v2f a; a.x = a1; a.y = a2;
      a12s[tid] = a; mskf[tid] = m ? 1.0f : 0.0f;
      size_t e = e0 + tid;
      out[OFF_EIDX + (size_t)b * 2 * ESZ + e]       = (float)srcRow;
      out[OFF_EIDX + (size_t)b * 2 * ESZ + ESZ + e] = (float)d;
      out[OFF_MASK + (size_t)b * ESZ + e]           = m ? 1.0f : 0.0f;
    } else {
      v2f z; z.x = 0.0f; z.y = 0.0f;
      a12s[tid] = z; mskf[tid] = 0.0f;
    }
  }
  __syncthreads();   // keys dead from here; shraw reused as embS

  // ---- emb = [a1 a2 1 0] x [w0; w1; bias; 0] via V_WMMA_F32_16X16X4_F32
  // Tile map: tN = waveId (B operand loop-invariant), tM = 0..6.
  // Branch-free operand build: unconditional loads + per-half value selects.
  const int  wave   = tid >> 5;         // 0..7 == tN
  const int  lane   = tid & 31;
  const int  m16    = lane & 15;
  const bool hiHalf = (lane >= 16);
  const int  col    = wave * 16 + m16;  // fixed per wave
  // B 4x16: lanes 0-15 -> (w0,w1); lanes 16-31 -> (bias,0)  [hoisted]
  const float w0c = edge_w[col];
  const float w1c = edge_w[DEMB + col];
  const float bc  = edge_b[col];
  v2f Bm;
  Bm.x = hiHalf ? bc   : w0c;
  Bm.y = hiHalf ? 0.0f : w1c;
  for (int tM = 0; tM < 7; ++tM) {
    // A 16x4: lanes 0-15 -> (K0,K1)=(a1,a2); lanes 16-31 -> (K2,K3)=(1,0)
    v2f a = a12s[tM * 16 + m16];        // one ds_load_b64, all lanes
    v2f A;
    A.x = hiHalf ? 1.0f : a.x;
    A.y = hiHalf ? 0.0f : a.y;
    v8f C = {0.f, 0.f, 0.f, 0.f, 0.f, 0.f, 0.f, 0.f};
    v8f D = __builtin_amdgcn_wmma_f32_16x16x4_f32(
        false, A, false, Bm, (short)0, C, false, false);
    int rbase = tM * 16 + (hiHalf ? 8 : 0);
#pragma unroll
    for (int v = 0; v < 8; ++v) {
      embS[(rbase + v) * DEMB + col] = D[v];   // D VGPR v -> matrix row M
    }
  }
  __syncthreads();

  // ---- masked, fully coalesced b128 writeback of the 100 valid rows
  const size_t embBase = OFF_EMB + ((size_t)b * ESZ + e0) * DEMB;
  v4f* __restrict__ outv = reinterpret_cast<v4f*>(out + embBase);
  const v4f* __restrict__ embv = reinterpret_cast<const v4f*>(embS);
  for (int i4 = tid; i4 < (KSP * DEMB) / 4; i4 += BLK) {
    int row = i4 >> 5;               // 32 float4 per 128-wide row
    v4f t = embv[i4];
    float s = mskf[row];
    t.x *= s; t.y *= s; t.z *= s; t.w *= s;
    outv[i4] = t;
  }
}

extern "C" void kernel_launch(void* const* d_in, const int* in_sizes, int n_in,
                              void* d_out, int out_size, void* d_ws, size_t ws_size,
                              hipStream_t stream) {
  const float* locs     = (const float*)d_in[0];
  const float* init_emb = (const float*)d_in[1];
  const float* edge_w   = (const float*)d_in[2];
  const float* edge_b   = (const float*)d_in[3];
  float* out = (float*)d_out;
  (void)in_sizes; (void)n_in; (void)d_ws; (void)ws_size; (void)out_size;

  // Output 1: init_embeddings passthrough (8 MB, D2D, graph-capture safe)
  hipMemcpyAsync(out, init_emb, (size_t)NB * NN * DEMB * sizeof(float),
                 hipMemcpyDeviceToDevice, stream);

  // Outputs 2-4: one block per selection problem (16 batches x 1001 rows)
  dim3 grid(NB * (NN + 1));
  vrp_edges_kernel<<<grid, BLK, 0, stream>>>(locs, edge_w, edge_b, out);
}